// SpectralAtomwise_27547920236675
// MI455X (gfx1250) — compile-verified
//
#include <hip/hip_runtime.h>
#include <stdint.h>

// SO(3) gated block-linear stack, fused into one kernel.
// out[n,r,:] = ( gate/silu( x[n,r,:] @ W1[l(r)] + b1*[r==0] ) ) @ W2[l(r)] + b2*[r==0]
// gates = sigmoid(silu(x[n,0,:] @ W0 + b0)), kept entirely in registers (lane-exact
// alignment with the WMMA C/D fragment layout).
// All global->LDS staging is done by the Tensor Data Mover (tensor_load_to_lds),
// double-buffered on the x row-tiles, synchronized with s_wait_tensorcnt + barriers.

typedef __attribute__((ext_vector_type(2))) float v2f;
typedef __attribute__((ext_vector_type(8))) float v8f;
typedef __attribute__((ext_vector_type(4))) unsigned int u32x4;
typedef __attribute__((ext_vector_type(8))) int i32x8;
typedef __attribute__((ext_vector_type(4))) int i32x4;

#define RTOT  49     // (LMAX+1)^2
#define CDIM  128    // SC == HC
#define GCDIM 768    // LMAX*HC
#define PITCH 132    // LDS row pitch in floats: TDM pads 4 DWORDs after every 128 DWORDs

__device__ __forceinline__ float fast_sigmoid(float v) {
  // single v_exp_f32 + v_rcp_f32 instead of the IEEE div_scale/fma chain
  return __builtin_amdgcn_rcpf(1.0f + __expf(-v));
}

__device__ __forceinline__ unsigned lds_off_u32(const void* p) {
  return (unsigned)(uintptr_t)p;   // low 32 bits of a generic LDS pointer = LDS byte offset
}

// Issue one TDM 2-D tile load: tile_d1 rows of tile_d0 f32 elements, source rows
// row_stride_elems apart, destination LDS padded to PITCH floats per row.
__device__ __forceinline__ void tdm_load_2d(unsigned lds_byte_off, const float* gptr,
                                            unsigned tens_d0, unsigned tens_d1,
                                            unsigned tile_d0, unsigned tile_d1,
                                            unsigned long long row_stride_elems) {
  const unsigned long long ga = (unsigned long long)(uintptr_t)gptr;
  u32x4 g0;
  g0[0] = 1u;                                                // count=1, no gather
  g0[1] = lds_byte_off;                                      // lds_addr [63:32]
  g0[2] = (unsigned)(ga & 0xffffffffull);                    // global_addr [95:64]
  g0[3] = (unsigned)((ga >> 32) & 0x1ffffffull) | (2u << 30);// addr[56:32] | type=2
  i32x8 g1;
  // data_size=4B(code2), pad_enable, pad_interval=128DW(code6), pad_amount=4DW(code3)
  g1[0] = (int)((2u << 16) | (1u << 20) | (6u << 22) | (3u << 25));
  g1[1] = (int)((tens_d0 & 0xffffu) << 16);                  // tensor_dim0 lo -> bits[63:48]
  g1[2] = (int)((tens_d0 >> 16) | ((tens_d1 & 0xffffu) << 16)); // dim0 hi | dim1 lo
  g1[3] = (int)((tens_d1 >> 16) | (tile_d0 << 16));          // dim1 hi | tile_dim0
  g1[4] = (int)(tile_d1 & 0xffffu);                          // tile_dim1 (tile_dim2 = 0)
  g1[5] = (int)(row_stride_elems & 0xffffffffull);           // tensor_dim0_stride lo
  g1[6] = (int)((row_stride_elems >> 32) & 0xffffull);       // stride hi | dim1_stride lo(0)
  g1[7] = 0;
  const i32x4 z4 = {0, 0, 0, 0};
  const i32x8 z8 = {0, 0, 0, 0, 0, 0, 0, 0};
  // this toolchain: 6-arg form (g0, g1, g2, g3, extra, cpol)
  __builtin_amdgcn_tensor_load_to_lds(g0, g1, z4, z4, z8, 0);
}

// acc += A(16x128 rows in aLds, pitch PITCH) * B(128x128 row-major [K][N] in bLds).
// WMMA f32 16x16x4 chained 32x along K.
// A frag: lane<16 -> (K=4k,4k+1) of row M=lane; lane>=16 -> (K=4k+2,4k+3).  B mirrors in K.
__device__ __forceinline__ v8f mm_k128(const float* __restrict__ aLds,
                                       const float* __restrict__ bLds,
                                       int m16, int colw, int hi, v8f acc) {
  #pragma unroll
  for (int k = 0; k < 32; ++k) {
    const int kr = 4 * k + 2 * hi;
    const v2f a = *(const v2f*)(aLds + m16 * PITCH + kr);
    v2f b;
    b.x = bLds[kr * PITCH + colw];
    b.y = bLds[(kr + 1) * PITCH + colw];
    acc = __builtin_amdgcn_wmma_f32_16x16x4_f32(false, a, false, b, (short)0, acc,
                                                false, false);
  }
  return acc;
}

__global__ __launch_bounds__(256, 1)
void so3_gated_mlp_kernel(const float* __restrict__ x,  const float* __restrict__ W0,
                          const float* __restrict__ b0, const float* __restrict__ W1,
                          const float* __restrict__ b1, const float* __restrict__ W2,
                          const float* __restrict__ b2, float* __restrict__ out, int N) {
  __shared__ float sW1[CDIM * PITCH];    // 66 KB, row-major [K][N], TDM-padded
  __shared__ float sW2[CDIM * PITCH];    // 66 KB
  __shared__ float xs[2][16 * PITCH];    // 2 x 8.25 KB, double-buffered x row-tiles
  __shared__ float hs[16 * PITCH];       // 8.25 KB

  const int tid  = threadIdx.x;
  const int wave = tid >> 5;             // 0..7 -> owns output cols [16w, 16w+16)
  const int lane = tid & 31;
  const int hi   = lane >> 4;
  const int m16  = lane & 15;
  const int colw = (wave << 4) + m16;    // this lane's output channel 0..127
  const int n0   = blockIdx.x << 4;      // 16 atoms per block
  const unsigned rows = (unsigned)((N - n0) < 16 ? (N - n0) : 16);
  const unsigned long long xstride = (unsigned long long)RTOT * CDIM;

  const float b1v = b1[colw];
  const float b2v = b2[colw];

  // ---------------- r=0 tile via TDM, then gating ----------------
  if (wave == 0) {
    tdm_load_2d(lds_off_u32(xs[0]), x + (size_t)n0 * RTOT * CDIM,
                CDIM, rows, CDIM, 16, xstride);
    __builtin_amdgcn_s_wait_tensorcnt(0);
  }
  __syncthreads();

  v8f gfr[6];
  #pragma unroll
  for (int j = 0; j < 6; ++j) {
    const int gcol = j * CDIM + colw;    // wave w handles gating tiles j*8+w exactly
    v8f acc = {0.f, 0.f, 0.f, 0.f, 0.f, 0.f, 0.f, 0.f};
    #pragma unroll 4
    for (int k = 0; k < 32; ++k) {
      const int kr = 4 * k + 2 * hi;
      const v2f a = *(const v2f*)(&xs[0][0] + m16 * PITCH + kr);
      v2f b;
      b.x = W0[(size_t)kr * GCDIM + gcol];
      b.y = W0[(size_t)(kr + 1) * GCDIM + gcol];
      acc = __builtin_amdgcn_wmma_f32_16x16x4_f32(false, a, false, b, (short)0, acc,
                                                  false, false);
    }
    const float b0v = b0[gcol];
    #pragma unroll
    for (int e = 0; e < 8; ++e) {
      const float t = acc[e] + b0v;
      gfr[j][e] = fast_sigmoid(t * fast_sigmoid(t));  // sigmoid(silu(.))
    }
  }

  // ---------------- main pass over degrees ----------------
  int r = 0, buf = 0;
  for (int l = 0; l <= 6; ++l) {
    __syncthreads();                     // all prior users of sW1/sW2/xs[buf] done
    if (wave == 0) {
      tdm_load_2d(lds_off_u32(sW1), W1 + (size_t)l * CDIM * CDIM, CDIM, CDIM, CDIM, CDIM, CDIM);
      tdm_load_2d(lds_off_u32(sW2), W2 + (size_t)l * CDIM * CDIM, CDIM, CDIM, CDIM, CDIM, CDIM);
      tdm_load_2d(lds_off_u32(xs[buf]), x + ((size_t)n0 * RTOT + r) * CDIM,
                  CDIM, rows, CDIM, 16, xstride);
    }
    const int nr = 2 * l + 1;
    for (int rr = 0; rr < nr; ++rr, ++r) {
      const bool hasnext = (rr + 1 < nr);
      if (wave == 0) {
        if (hasnext) {                   // prefetch next row-tile into the other buffer
          tdm_load_2d(lds_off_u32(xs[buf ^ 1]), x + ((size_t)n0 * RTOT + r + 1) * CDIM,
                      CDIM, rows, CDIM, 16, xstride);
          __builtin_amdgcn_s_wait_tensorcnt(1);  // in-order: weights + current tile done
        } else {
          __builtin_amdgcn_s_wait_tensorcnt(0);
        }
      }
      __syncthreads();                   // release consumers; staged data visible

      // h = x_r @ W1[l]
      v8f h = {0.f, 0.f, 0.f, 0.f, 0.f, 0.f, 0.f, 0.f};
      h = mm_k128(xs[buf], sW1, m16, colw, hi, h);

      if (r == 0) {
        #pragma unroll
        for (int e = 0; e < 8; ++e) { const float t = h[e] + b1v; h[e] = t * fast_sigmoid(t); }
      } else {
        const v8f g = gfr[l - 1];
        #pragma unroll
        for (int e = 0; e < 8; ++e) h[e] *= g[e];
      }

      // publish h tile for the second block-linear
      #pragma unroll
      for (int v = 0; v < 8; ++v) hs[(v + (hi << 3)) * PITCH + colw] = h[v];
      __syncthreads();

      // o = h @ W2[l]
      v8f o = {0.f, 0.f, 0.f, 0.f, 0.f, 0.f, 0.f, 0.f};
      o = mm_k128(hs, sW2, m16, colw, hi, o);
      if (r == 0) {
        #pragma unroll
        for (int e = 0; e < 8; ++e) o[e] += b2v;
      }

      const size_t ob = ((size_t)n0 * RTOT + r) * CDIM;
      #pragma unroll
      for (int v = 0; v < 8; ++v) {
        const int mrow = v + (hi << 3);
        if (n0 + mrow < N) out[ob + (size_t)mrow * RTOT * CDIM + colw] = o[v];
      }
      if (hasnext) buf ^= 1;
    }
  }
}

extern "C" void kernel_launch(void* const* d_in, const int* in_sizes, int n_in,
                              void* d_out, int out_size, void* d_ws, size_t ws_size,
                              hipStream_t stream) {
  const float* x  = (const float*)d_in[0];
  const float* W0 = (const float*)d_in[1];
  const float* b0 = (const float*)d_in[2];
  const float* W1 = (const float*)d_in[3];
  const float* b1 = (const float*)d_in[4];
  const float* W2 = (const float*)d_in[5];
  const float* b2 = (const float*)d_in[6];
  float* out = (float*)d_out;

  const int N = in_sizes[0] / (RTOT * CDIM);           // 10000
  dim3 grid((N + 15) / 16), block(256);
  so3_gated_mlp_kernel<<<grid, block, 0, stream>>>(x, W0, b0, W1, b1, W2, b2, out, N);
}